// CovarianceLayer_1580547967699
// MI455X (gfx1250) — compile-verified
//
#include <hip/hip_runtime.h>
#include <cstdint>

#ifndef __has_builtin
#define __has_builtin(x) 0
#endif

typedef float v2f __attribute__((ext_vector_type(2)));
typedef float v8f __attribute__((ext_vector_type(8)));

#define AS1 __attribute__((address_space(1)))
#define AS3 __attribute__((address_space(3)))

static __device__ __forceinline__ int imin(int a, int b) { return a < b ? a : b; }

// ---- Async global->LDS copy (gfx1250 GLOBAL_LOAD_ASYNC_TO_LDS_B32, ASYNCcnt) ----
static __device__ __forceinline__ void g2lds_b32(const float* g, float* l) {
#if __has_builtin(__builtin_amdgcn_global_load_async_to_lds_b32)
  __builtin_amdgcn_global_load_async_to_lds_b32(
      (AS1 int*)(uintptr_t)g,
      (AS3 int*)(unsigned)(uintptr_t)l,
      0, 0);
#else
  *l = *g;  // sync fallback (still compiles; async path preferred)
#endif
}

static __device__ __forceinline__ void wait_g2lds() {
#if __has_builtin(__builtin_amdgcn_global_load_async_to_lds_b32)
#if __has_builtin(__builtin_amdgcn_s_wait_asynccnt)
  __builtin_amdgcn_s_wait_asynccnt(0);
#else
  asm volatile("s_wait_asynccnt 0" ::: "memory");
#endif
#endif
}

// ---- Problem geometry ----
constexpr int IN_W  = 1024;   // input side
constexpr int OUT_W = 1016;   // output side (valid 5x5 twice)
constexpr int TO    = 48;     // output tile side
constexpr int TI    = TO + 8; // 56: input region (halo 4+4)
constexpr int TD    = TO + 4; // 52: dev/product region
constexpr int STR   = 58;     // LDS row stride (floats)
constexpr int NTILE = (OUT_W + TO - 1) / TO;  // 22
constexpr int NSUB  = TO / 16;                // 3 WMMA subtiles per dim

__global__ __launch_bounds__(256)
void cov5x5_wmma_kernel(const float* __restrict__ x,
                        const float* __restrict__ y,
                        float* __restrict__ out) {
  __shared__ float xs[TI * STR];  // x tile; later aliased as vp
  __shared__ float ys[TI * STR];  // y tile
  __shared__ float vx[TD * STR];  // vertical 5-sum of x
  __shared__ float vy[TD * STR];  // vertical 5-sum of y
  __shared__ float pp[TD * STR];  // dev_x * dev_y

  const int tid = threadIdx.x;
  const int b   = blockIdx.y;
  const int ty  = blockIdx.x / NTILE;
  const int tx  = blockIdx.x % NTILE;
  const int oy  = ty * TO;
  const int ox  = tx * TO;

  const float* xb = x + ((size_t)b << 20);
  const float* yb = y + ((size_t)b << 20);

  // ---- Stage 0: async-stage 56x56 regions of x and y into LDS (clamped) ----
  for (int i = tid; i < TI * TI; i += 256) {
    const int r  = i / TI, c = i % TI;
    const int gr = imin(oy + r, IN_W - 1);
    const int gc = imin(ox + c, IN_W - 1);
    const size_t go = ((size_t)gr << 10) + (size_t)gc;
    g2lds_b32(xb + go, &xs[r * STR + c]);
    g2lds_b32(yb + go, &ys[r * STR + c]);
  }
  wait_g2lds();
  __syncthreads();

  // ---- Stage A: vertical 5-sums of x and y (52 x 56) ----
  for (int i = tid; i < TD * TI; i += 256) {
    const int r = i / TI, c = i % TI;
    const int o = r * STR + c;
    vx[o] = xs[o] + xs[o + STR] + xs[o + 2 * STR] + xs[o + 3 * STR] + xs[o + 4 * STR];
    vy[o] = ys[o] + ys[o + STR] + ys[o + 2 * STR] + ys[o + 3 * STR] + ys[o + 4 * STR];
  }
  __syncthreads();

  // ---- Stage B: dev product p = (xc - mean(x)) * (yc - mean(y)) (52 x 52) ----
  constexpr float inv25 = 1.0f / 25.0f;
  for (int i = tid; i < TD * TD; i += 256) {
    const int r = i / TD, c = i % TD;
    const int o = r * STR + c;
    const float mx = (vx[o] + vx[o + 1] + vx[o + 2] + vx[o + 3] + vx[o + 4]) * inv25;
    const float my = (vy[o] + vy[o + 1] + vy[o + 2] + vy[o + 3] + vy[o + 4]) * inv25;
    const float dx = xs[(r + 2) * STR + (c + 2)] - mx;
    const float dy = ys[(r + 2) * STR + (c + 2)] - my;
    pp[o] = dx * dy;
  }
  __syncthreads();

  // ---- Stage C: vertical 5-sum of p -> vp (48 x 52), aliases xs ----
  float* vp = xs;  // xs is dead from here on
  for (int i = tid; i < TO * TD; i += 256) {
    const int r = i / TD, c = i % TD;
    const int o = r * STR + c;
    vp[o] = pp[o] + pp[o + STR] + pp[o + 2 * STR] + pp[o + 3 * STR] + pp[o + 4 * STR];
  }
  __syncthreads();

  // ---- Stage D: horizontal 5-sum via WMMA band-matrix matmul ----
  // D(16x16) = VP(16x20) x S(20x16), S[k][n] = 1 iff n <= k <= n+4,
  // chained as 5 x V_WMMA_F32_16X16X4_F32 accumulations.
  const int wave = tid >> 5;
  const int lane = tid & 31;
  const int m = lane & 15;   // A: row M; B: col N; D: col N
  const int h = lane >> 4;   // half-wave select for K / M+8

  for (int s = wave; s < NSUB * NSUB; s += 8) {   // wave-uniform trip count
    const int tr = (s / NSUB) * 16;
    const int tc = (s % NSUB) * 16;
    v8f acc = {};
#pragma unroll
    for (int kc = 0; kc < 5; ++kc) {
      const int k0 = 4 * kc + 2 * h;  // K indices held by this lane: k0, k0+1
      v2f a, bb;
      a.x = vp[(tr + m) * STR + tc + k0];
      a.y = vp[(tr + m) * STR + tc + k0 + 1];
      bb.x = (k0     >= m && k0     <= m + 4) ? 1.0f : 0.0f;
      bb.y = (k0 + 1 >= m && k0 + 1 <= m + 4) ? 1.0f : 0.0f;
      acc = __builtin_amdgcn_wmma_f32_16x16x4_f32(
          false, a, false, bb, (short)0, acc, false, false);
    }
    // D layout: VGPR v -> row M = v + 8*h, col N = m
#pragma unroll
    for (int v = 0; v < 8; ++v) {
      const int orow = oy + tr + v + 8 * h;
      const int ocol = ox + tc + m;
      if (orow < OUT_W && ocol < OUT_W) {
        out[((size_t)b * OUT_W + orow) * OUT_W + ocol] = acc[v] * inv25;
      }
    }
  }
}

extern "C" void kernel_launch(void* const* d_in, const int* in_sizes, int n_in,
                              void* d_out, int out_size, void* d_ws, size_t ws_size,
                              hipStream_t stream) {
  const float* x = (const float*)d_in[0];
  const float* y = (const float*)d_in[1];
  // d_in[2] is mean_mask (constant 1/25), folded into the kernel.
  float* out = (float*)d_out;

  dim3 grid(NTILE * NTILE, 16);
  cov5x5_wmma_kernel<<<grid, 256, 0, stream>>>(x, y, out);
}